// TAGClassifier_69999376990328
// MI455X (gfx1250) — compile-verified
//
#include <hip/hip_runtime.h>

typedef __attribute__((ext_vector_type(2))) float v2f;
typedef __attribute__((ext_vector_type(8))) float v8f;

#define N_NODES 100000
#define N_EDGES 3200000
#define IN_DIM  128
#define HID     256
#define NCLS    10
#define NGRAPH  64

// ---------------------------------------------------------------- utilities

__global__ void k_zero4(float4* __restrict__ p, long n4) {
  long i = (long)blockIdx.x * blockDim.x + threadIdx.x;
  if (i < n4) {
    float4 z; z.x = 0.f; z.y = 0.f; z.z = 0.f; z.w = 0.f;
    p[i] = z;
  }
}

__global__ void k_deg(const int* __restrict__ dst, float* __restrict__ deg) {
  int i = blockIdx.x * blockDim.x + threadIdx.x;
  if (i < N_EDGES) atomicAdd(&deg[dst[i]], 1.0f);
}

__global__ void k_norm(float* __restrict__ norm) {
  int i = blockIdx.x * blockDim.x + threadIdx.x;
  if (i < N_NODES) {
    float d = norm[i];
    d = d < 1.0f ? 1.0f : d;
    norm[i] = 1.0f / sqrtf(d);
  }
}

// One wave (d=128) or two waves (d=256) per edge (chunks of 4 floats per lane): src[e]/dst[e] are
// wave-uniform (scalar broadcast loads), feature reads are 512B/1KB
// contiguous. Node features live in L2 (192MB), so gather+atomics stay on-chip.
__global__ void k_scatter(const float* __restrict__ cur, const float* __restrict__ norm,
                          const int* __restrict__ src, const int* __restrict__ dst,
                          float* __restrict__ agg, int log2c) {
  long idx = (long)blockIdx.x * blockDim.x + threadIdx.x;
  long total = (long)N_EDGES << log2c;
  if (idx >= total) return;
  int e = (int)(idx >> log2c);
  int c = (int)(idx & ((1 << log2c) - 1));
  int s = src[e];
  int t = dst[e];
  float ns = norm[s];
  int d = 4 << log2c;
  const float4 v = *((const float4*)(cur + (long)s * d) + c);
  float* o = agg + (long)t * d + 4 * c;
  atomicAdd(o + 0, v.x * ns);
  atomicAdd(o + 1, v.y * ns);
  atomicAdd(o + 2, v.z * ns);
  atomicAdd(o + 3, v.w * ns);
}

__global__ void k_scale(float4* __restrict__ x, const float* __restrict__ norm, int log2c) {
  long idx = (long)blockIdx.x * blockDim.x + threadIdx.x;
  long total = (long)N_NODES << log2c;
  if (idx >= total) return;
  long row = idx >> log2c;
  float s = norm[row];
  float4 v = x[idx];
  v.x *= s; v.y *= s; v.z *= s; v.w *= s;
  x[idx] = v;
}

// ------------------------------------------------------- fused TAGConv GEMM
// out[N_NODES x 256] = relu( [A0|A1|A2] (N x 3d) @ W (3d x 256) + bias )
// One wave32 computes a 16x64 output strip with V_WMMA_F32_16X16X4_F32,
// reusing each A fragment across 4 N-tiles. mode 1 fuses segment-max pooling
// (atomicMax on float bits; valid because relu output >= 0).
__global__ __launch_bounds__(256)
void k_tag_gemm(const float* __restrict__ A0, const float* __restrict__ A1,
                const float* __restrict__ A2, int d,
                const float* __restrict__ W, const float* __restrict__ bias,
                float* __restrict__ out, float* __restrict__ pooled,
                const int* __restrict__ gids, int mode) {
  const int NOUT = HID; // 256
  int gwave = (int)(((long)blockIdx.x * blockDim.x + threadIdx.x) >> 5);
  int lane  = threadIdx.x & 31;
  int tile_m  = gwave >> 2;   // 256/64 = 4 n-strips
  int tile_nb = gwave & 3;
  int m0 = tile_m * 16;       // M = 100000 = 6250*16, exact
  int n0 = tile_nb * 64;
  int half = lane >> 4;       // A/B fragment: lanes 16-31 hold K+2
  int mrow = lane & 15;
  int kk   = half * 2;

  v8f c0 = {}, c1 = {}, c2 = {}, c3 = {};
  const float* parts[3] = {A0, A1, A2};
  for (int p = 0; p < 3; ++p) {
    const float* arow = parts[p] + (long)(m0 + mrow) * d;
    const float* Wp   = W + (long)p * d * NOUT;
    for (int k0 = 0; k0 < d; k0 += 4) {
      v2f a;
      a.x = arow[k0 + kk];
      a.y = arow[k0 + kk + 1];
      const float* w0 = Wp + (long)(k0 + kk) * NOUT + n0 + mrow;
      const float* w1 = w0 + NOUT;
      v2f b;
      b.x = w0[0];  b.y = w1[0];
      c0 = __builtin_amdgcn_wmma_f32_16x16x4_f32(false, a, false, b, (short)0, c0, false, false);
      b.x = w0[16]; b.y = w1[16];
      c1 = __builtin_amdgcn_wmma_f32_16x16x4_f32(false, a, false, b, (short)0, c1, false, false);
      b.x = w0[32]; b.y = w1[32];
      c2 = __builtin_amdgcn_wmma_f32_16x16x4_f32(false, a, false, b, (short)0, c2, false, false);
      b.x = w0[48]; b.y = w1[48];
      c3 = __builtin_amdgcn_wmma_f32_16x16x4_f32(false, a, false, b, (short)0, c3, false, false);
    }
  }

  v8f acc[4] = {c0, c1, c2, c3};
  // C/D layout: VGPR r -> M = m0 + r (+8 for lanes 16-31), N = n0 + (lane&15)
  for (int j = 0; j < 4; ++j) {
    int col = n0 + j * 16 + mrow;
    float bb = bias[col];
#pragma unroll
    for (int r = 0; r < 8; ++r) {
      int row = m0 + r + half * 8;
      float v = acc[j][r] + bb;
      v = v > 0.f ? v : 0.f;
      if (mode == 0) {
        out[(long)row * NOUT + col] = v;
      } else {
        atomicMax((int*)(pooled + (long)gids[row] * NOUT + col), __float_as_int(v));
      }
    }
  }
}

// -------------------------------------------------------- tiny classifier
__global__ void k_final(const float* __restrict__ pooled, const float* __restrict__ Wc,
                        const float* __restrict__ bc, float* __restrict__ out) {
  int g = blockIdx.x;
  int c = threadIdx.x;
  if (c < NCLS) {
    float acc = bc[c];
    const float* pr = pooled + g * HID;
    for (int k = 0; k < HID; ++k) acc += pr[k] * Wc[k * NCLS + c];
    out[g * NCLS + c] = acc;
  }
}

// ---------------------------------------------------------------- launcher

extern "C" void kernel_launch(void* const* d_in, const int* in_sizes, int n_in,
                              void* d_out, int out_size, void* d_ws, size_t ws_size,
                              hipStream_t stream) {
  (void)in_sizes; (void)n_in; (void)out_size; (void)ws_size;
  const float* h   = (const float*)d_in[0];
  const int*   src = (const int*)d_in[1];
  const int*   dst = (const int*)d_in[2];
  const int*   gid = (const int*)d_in[3];
  const float* W1  = (const float*)d_in[4];
  const float* b1  = (const float*)d_in[5];
  const float* W2  = (const float*)d_in[6];
  const float* b2  = (const float*)d_in[7];
  const float* Wc  = (const float*)d_in[8];
  const float* bc  = (const float*)d_in[9];
  float* out = (float*)d_out;

  char* base = (char*)d_ws;
  size_t off = 0;
  auto take = [&](size_t bytes) {
    char* p = base + off;
    off += (bytes + 255) & ~(size_t)255;
    return p;
  };
  float* norm   = (float*)take((size_t)N_NODES * 4);
  float* hopAB  = (float*)take((size_t)N_NODES * 256 * 4); // L1 hop1+hop2 / L2 hop1
  float* x1     = (float*)take((size_t)N_NODES * 256 * 4);
  float* hopC   = (float*)take((size_t)N_NODES * 256 * 4); // L2 hop2
  float* pooled = (float*)take((size_t)NGRAPH * 256 * 4);

  float* h1a = hopAB;
  float* h1b = hopAB + (size_t)N_NODES * IN_DIM;
  float* h2a = hopAB;  // reuse: layer-1 hops dead after GEMM1
  float* h2b = hopC;

  const int TB = 256;
  auto blocks = [](long n) { return (unsigned)((n + 255) / 256); };

  // degree -> norm = deg^-1/2 (clipped)
  k_zero4<<<blocks(N_NODES / 4), TB, 0, stream>>>((float4*)norm, (long)(N_NODES / 4));
  k_deg  <<<blocks(N_EDGES), TB, 0, stream>>>(dst, norm);
  k_norm <<<blocks(N_NODES), TB, 0, stream>>>(norm);

  // ---- layer 1 (d = 128, log2chunks = 5) ----
  long n4_128 = (long)N_NODES * (IN_DIM / 4);
  long sc128  = (long)N_EDGES << 5;
  k_zero4  <<<blocks(n4_128), TB, 0, stream>>>((float4*)h1a, n4_128);
  k_scatter<<<blocks(sc128), TB, 0, stream>>>(h, norm, src, dst, h1a, 5);
  k_scale  <<<blocks(n4_128), TB, 0, stream>>>((float4*)h1a, norm, 5);
  k_zero4  <<<blocks(n4_128), TB, 0, stream>>>((float4*)h1b, n4_128);
  k_scatter<<<blocks(sc128), TB, 0, stream>>>(h1a, norm, src, dst, h1b, 5);
  k_scale  <<<blocks(n4_128), TB, 0, stream>>>((float4*)h1b, norm, 5);

  // x1 = relu([h|h1a|h1b] @ W1 + b1): 6250*4 waves = 3125 blocks of 8 waves
  const unsigned GEMM_BLOCKS = (N_NODES / 16) * (HID / 64) / 8; // 3125
  k_tag_gemm<<<GEMM_BLOCKS, TB, 0, stream>>>(h, h1a, h1b, IN_DIM, W1, b1,
                                             x1, nullptr, nullptr, 0);

  // ---- layer 2 (d = 256, log2chunks = 6) ----
  long n4_256 = (long)N_NODES * (HID / 4);
  long sc256  = (long)N_EDGES << 6;
  k_zero4  <<<blocks(n4_256), TB, 0, stream>>>((float4*)h2a, n4_256);
  k_scatter<<<blocks(sc256), TB, 0, stream>>>(x1, norm, src, dst, h2a, 6);
  k_scale  <<<blocks(n4_256), TB, 0, stream>>>((float4*)h2a, norm, 6);
  k_zero4  <<<blocks(n4_256), TB, 0, stream>>>((float4*)h2b, n4_256);
  k_scatter<<<blocks(sc256), TB, 0, stream>>>(h2a, norm, src, dst, h2b, 6);
  k_scale  <<<blocks(n4_256), TB, 0, stream>>>((float4*)h2b, norm, 6);

  // pooled = segment_max(relu([x1|h2a|h2b] @ W2 + b2)) fused into GEMM epilogue
  k_zero4<<<blocks(NGRAPH * HID / 4), TB, 0, stream>>>((float4*)pooled,
                                                       (long)(NGRAPH * HID / 4));
  k_tag_gemm<<<GEMM_BLOCKS, TB, 0, stream>>>(x1, h2a, h2b, HID, W2, b2,
                                             nullptr, pooled, gid, 1);

  // out = pooled @ Wc + bc
  k_final<<<NGRAPH, 32, 0, stream>>>(pooled, Wc, bc, out);
}